// AC_13907104104944
// MI455X (gfx1250) — compile-verified
//
#include <hip/hip_runtime.h>

typedef __attribute__((ext_vector_type(2))) float v2f;
typedef __attribute__((ext_vector_type(8))) float v8f;

__device__ __forceinline__ float reluf(float x) { return fmaxf(x, 0.0f); }

__device__ __forceinline__ float rerelu(float x) {
    const float offs[24] = {3.25f, 3.75f, 5.25f, 5.75f, 6.25f, 6.75f, 11.25f, 11.75f,
                            15.25f, 15.75f, 20.25f, 20.75f, 23.25f, 23.75f, 35.25f, 35.75f,
                            40.25f, 40.75f, 43.25f, 43.75f, 49.25f, 49.75f, 61.25f, 61.75f};
    const float coefs[24] = {4.f, -4.f, -4.f, 4.f, 4.f, -4.f, -2.f, 2.f,
                             2.f, -2.f, -2.f, 2.f, 2.f, -2.f, -2.f, 2.f,
                             2.f, -2.f, -2.f, 2.f, 2.f, -2.f, -2.f, 2.f};
    float acc = 0.0f;
#pragma unroll
    for (int k = 0; k < 24; ++k) acc = fmaf(coefs[k], reluf(x - offs[k]), acc);
    return acc;
}

__device__ __forceinline__ float rerelu2(float x) {
    const float offs[6]  = {1.25f, 1.75f, 2.25f, 2.75f, 3.25f, 3.75f};
    const float coefs[6] = {-2.f, 2.f, 2.f, -2.f, -2.f, 2.f};
    float acc = 1.0f;
#pragma unroll
    for (int k = 0; k < 6; ++k) acc = fmaf(coefs[k], reluf(x - offs[k]), acc);
    return acc;
}

__device__ __forceinline__ float rerelu3(float x) {
    return 2.0f * reluf(x - 1.25f) - 2.0f * reluf(x - 1.75f);
}

__device__ __forceinline__ float rerelu4(float x) {
    // offs [0,1,1,21], coefs [8,-8,-0.3,0.3]
    return 8.0f * reluf(x) - 8.0f * reluf(x - 1.0f)
         - 0.3f * reluf(x - 1.0f) + 0.3f * reluf(x - 21.0f);
}

// Kernel 1: per-token table (start, end, nt^2) + frame->token scatter map.
// Single block, 256 threads, LDS exclusive scan of chunk sums.
__global__ __launch_bounds__(256)
void AC_tok_scan(const int* __restrict__ duration, const float* __restrict__ token,
                 float4* __restrict__ tab, int* __restrict__ ftok, int T) {
    __shared__ int csum[256];
    const int tid = threadIdx.x;
    const int C  = (T + 255) >> 8;          // tokens per thread chunk
    const int t0 = tid * C;
    const int t1 = min(t0 + C, T);

    int s = 0;
    for (int t = t0; t < t1; ++t) s += duration[t];
    csum[tid] = s;
    __syncthreads();
    if (tid == 0) {                          // tiny serial exclusive scan (256 entries)
        int run = 0;
        for (int i = 0; i < 256; ++i) { int v = csum[i]; csum[i] = run; run += v; }
    }
    __syncthreads();

    int start = csum[tid];
    for (int t = t0; t < t1; ++t) {
        const int   d  = duration[t];
        const float df = (float)d;
        const float tk = rerelu(token[t]);
        const float tk1 = (t + 1 < T) ? rerelu(token[t + 1]) : 0.0f;  // F.pad([-1,1]) zero-fill
        const float dn  = (t + 1 < T) ? (float)duration[t + 1] : 0.0f;
        const float new_dur = fmaf(rerelu2(tk + tk1), dn, df);
        const float nt = rerelu3(tk);
        tab[t] = make_float4((float)start, (float)start + new_dur, nt * nt, 0.0f);
        for (int j = 0; j < d; ++j) ftok[start + j] = t;   // partitions [0, L)
        start += d;
    }
}

// Kernel 2: per-frame gather (<=2 covering tokens), rerelu4, then a WMMA
// f32 16x16x4 outer-product (B = const 0.25, A dense = val) to broadcast each
// wave's 16 frame values across the 16x16 tile, stored 8x across 128 channels.
__global__ __launch_bounds__(256)
void AC_mel(const float4* __restrict__ tab, const int* __restrict__ ftok,
            float* __restrict__ out, int L) {
    const int lane = threadIdx.x & 31;
    const int wave = threadIdx.x >> 5;
    const int frameBase = (blockIdx.x * (blockDim.x >> 5) + wave) << 4;  // 16 frames/wave
    const int l  = frameBase + (lane & 15);           // lanes m and m+16 share a frame
    const int lc = min(l, L - 1);                     // clamp for safe reads; no early exit

    const int t = ftok[lc];
    const float4 q = tab[t];
    const float lf = (float)lc;
    float acc = q.z * (lf - q.x + 1.0f);              // nt2 * (l - start + 1)
    if (t > 0) {
        const float4 p = tab[t - 1];
        if (lf < p.y) acc = fmaf(p.z, lf - p.x + 1.0f, acc);
    }
    const float v = rerelu4(acc);

    // Outer-product broadcast: A[m,k] = val[m] (all k), B[k,n] = 0.25 (all k,n)
    // => D[m,n] = 4 * val[m] * 0.25 = val[m]. Exact in f32, layout-robust.
    v2f a; a.x = v;      a.y = v;
    v2f b; b.x = 0.25f;  b.y = 0.25f;
    v8f c = {};
    v8f d = __builtin_amdgcn_wmma_f32_16x16x4_f32(false, a, false, b,
                                                  (short)0, c, false, false);

    // D layout: VGPR r -> row M = r (lanes 0-15) / r+8 (lanes 16-31), col N = lane&15.
    const int col  = lane & 15;
    const int half = (lane >> 4) << 3;  // 0 or 8
    float* base = out + (size_t)(frameBase + half) * 128 + col;

    if (frameBase + 16 <= L) {
        // Wave-uniform fast path: all 16 frames in range -> 64 unconditional
        // stores, no per-row exec-mask manipulation.
#pragma unroll
        for (int r = 0; r < 8; ++r) {
            const float val = d[r];
            float* row = base + (size_t)r * 128;
#pragma unroll
            for (int rep = 0; rep < 8; ++rep) row[rep * 16] = val;  // 8 x 16-ch replicas
        }
    } else {
        // Tail wave: per-row bounds check.
#pragma unroll
        for (int r = 0; r < 8; ++r) {
            const int f = frameBase + r + half;
            if (f < L) {
                const float val = d[r];
                float* row = out + (size_t)f * 128 + col;
#pragma unroll
                for (int rep = 0; rep < 8; ++rep) row[rep * 16] = val;
            }
        }
    }
}

extern "C" void kernel_launch(void* const* d_in, const int* in_sizes, int n_in,
                              void* d_out, int out_size, void* d_ws, size_t ws_size,
                              hipStream_t stream) {
    const int*   duration = (const int*)d_in[0];
    const float* token    = (const float*)d_in[1];
    // d_in[2] (f0) unused, matching the reference.
    const int T = in_sizes[0];
    const int L = out_size / 128;   // output is [1, L, 128] f32

    float4* tab  = (float4*)d_ws;                                   // T * 16 B
    int*    ftok = (int*)((char*)d_ws + (size_t)T * sizeof(float4)); // L * 4 B

    AC_tok_scan<<<1, 256, 0, stream>>>(duration, token, tab, ftok, T);

    const int framesPerBlock = 128;  // 8 waves * 16 frames
    const int grid = (L + framesPerBlock - 1) / framesPerBlock;
    AC_mel<<<grid, 256, 0, stream>>>(tab, ftok, (float*)d_out, L);
}